// DistributionNMS_55705725829814
// MI455X (gfx1250) — compile-verified
//
#include <hip/hip_runtime.h>
#include <stdint.h>

#define BATCH   16
#define NBOX    32768
#define NCLS    80
#define MAX_DET 200
#define IOU_THR 0.5f
#define SCORE_THR 0.01f
#define NEG_INF (-__builtin_inff())

typedef unsigned int u32x4_t __attribute__((ext_vector_type(4)));
typedef int          i32x8_t __attribute__((ext_vector_type(8)));
typedef int          i32x4_t __attribute__((ext_vector_type(4)));

// ---------------------------------------------------------------------------
// Pass A: one thread per (b,n) row.
//  - softmax stats over 80 logits -> masked score (score>thr ? score : -inf)
//  - canonicalized xywh box (a0,a1,b0,b1) as float4 (reference's _canon on xywh)
// Streaming, memory-bound: ~176MB read -> ~7.5us at 23.3 TB/s.
// ---------------------------------------------------------------------------
__global__ void score_canon_kernel(const float* __restrict__ boxp,
                                   const float* __restrict__ clsp,
                                   float* __restrict__ ms,
                                   float4* __restrict__ canon) {
  int r = blockIdx.x * blockDim.x + threadIdx.x;
  if (r >= BATCH * NBOX) return;

  const float4* cp4 = (const float4*)(clsp + (size_t)r * NCLS);  // 320B row, 16B aligned
  float4 v[NCLS / 4];
  float m = NEG_INF;
#pragma unroll
  for (int i = 0; i < NCLS / 4; ++i) {
    v[i] = cp4[i];
    m = fmaxf(m, fmaxf(fmaxf(v[i].x, v[i].y), fmaxf(v[i].z, v[i].w)));
  }
  float s = 0.f;
#pragma unroll
  for (int i = 0; i < NCLS / 4; ++i) {
    s += __expf(v[i].x - m) + __expf(v[i].y - m) +
         __expf(v[i].z - m) + __expf(v[i].w - m);
  }
  float score = 1.0f / s;  // max softmax prob = exp(m-m)/sum
  ms[r] = (score > SCORE_THR) ? score : NEG_INF;

  float4 bb = ((const float4*)boxp)[r];  // x1 y1 x2 y2
  float x = bb.x, y = bb.y, w = bb.z - bb.x, h = bb.w - bb.y;  // xywh as reference
  canon[r] = make_float4(fminf(x, w), fmaxf(x, w), fminf(y, h), fmaxf(y, h));
}

// ---------------------------------------------------------------------------
// Pass B: greedy NMS, one workgroup (32 wave32's) per image.
// Score row (128KB) lives in LDS (CDNA5: 320KB/WGP), loaded via TDM DMA.
// Each round fuses IoU suppression with the next round's local argmax scan.
// ---------------------------------------------------------------------------
__global__ __launch_bounds__(1024) void nms_kernel(const float* __restrict__ ms,
                                                   const float4* __restrict__ canon,
                                                   int* __restrict__ sel) {
  __shared__ float s_sc[NBOX];     // 128 KB masked scores
  __shared__ float s_rv[32];
  __shared__ int   s_ri[32];
  __shared__ float s_bv;
  __shared__ int   s_bi;
  __shared__ int   s_sel[MAX_DET];

  const int b = blockIdx.x;
  const int t = threadIdx.x;
  const float* row = ms + (size_t)b * NBOX;

  for (int i = t; i < MAX_DET; i += 1024) s_sel[i] = -1;

#if __has_builtin(__builtin_amdgcn_tensor_load_to_lds)
  if (t < 32) {  // wave 0 only: TDM ignores EXEC, other waves scalar-skip
    uint64_t ga  = (uint64_t)(uintptr_t)row;
    uint32_t lds = (uint32_t)(uintptr_t)&s_sc[0];  // LDS aperture: low 32 bits = offset
    u32x4_t g0;
    g0[0] = 1u;                                    // count = 1 valid descriptor
    g0[1] = lds;                                   // lds_addr
    g0[2] = (uint32_t)ga;                          // global_addr[31:0]
    g0[3] = (uint32_t)((ga >> 32) & 0x01FFFFFFu)   // global_addr[56:32]
          | (2u << 30);                            // type = 2 ("image")
    i32x8_t g1;
    g1[0] = (2 << 16);                             // data_size = 4 bytes
    g1[1] = (int)(((unsigned)NBOX & 0xFFFFu) << 16);    // tensor_dim0[15:0]
    g1[2] = (int)(((unsigned)NBOX >> 16) | (1u << 16)); // dim0[31:16] | tensor_dim1=1
    g1[3] = (int)(((unsigned)NBOX & 0xFFFFu) << 16);    // tile_dim0 = 32768
    g1[4] = 1;                                     // tile_dim1 = 1
    g1[5] = NBOX;                                  // tensor_dim0_stride[31:0]
    g1[6] = 0;
    g1[7] = 0;
    i32x4_t z4 = {0, 0, 0, 0};                     // groups 2/3 unused (2D tensor)
    i32x8_t z8 = {0, 0, 0, 0, 0, 0, 0, 0};
    // amdgpu-toolchain (clang-23) 6-arg form: (g0, g1, g2, g3, g_extra, cpol)
    __builtin_amdgcn_tensor_load_to_lds(g0, g1, z4, z4, z8, 0);
#if __has_builtin(__builtin_amdgcn_s_wait_tensorcnt)
    __builtin_amdgcn_s_wait_tensorcnt(0);
#else
    asm volatile("s_wait_tensorcnt 0" ::: "memory");
#endif
  }
#else
  for (int k = 0; k < NBOX / 1024; ++k) {
    int j = t + (k << 10);
    s_sc[j] = row[j];
  }
#endif
  __syncthreads();

  // initial local argmax over this thread's strided 32 entries
  float lv = NEG_INF;
  int   li = t;
  for (int k = 0; k < NBOX / 1024; ++k) {
    int j = t + (k << 10);
    float v2 = s_sc[j];
    if (v2 > lv) { lv = v2; li = j; }
  }

  const float4* cb = canon + (size_t)b * NBOX;
  const int wid = t >> 5, ln = t & 31;

  for (int it = 0; it < MAX_DET; ++it) {
    // ---- cross-workgroup argmax: wave32 shuffle, then one wave over 32 waves
    float v = lv; int ix = li;
#pragma unroll
    for (int o = 16; o > 0; o >>= 1) {
      float ov = __shfl_xor(v, o, 32);
      int   oi = __shfl_xor(ix, o, 32);
      if (ov > v || (ov == v && oi < ix)) { v = ov; ix = oi; }
    }
    if (ln == 0) { s_rv[wid] = v; s_ri[wid] = ix; }
    __syncthreads();
    if (wid == 0) {
      v = s_rv[ln]; ix = s_ri[ln];
#pragma unroll
      for (int o = 16; o > 0; o >>= 1) {
        float ov = __shfl_xor(v, o, 32);
        int   oi = __shfl_xor(ix, o, 32);
        if (ov > v || (ov == v && oi < ix)) { v = ov; ix = oi; }
      }
      if (ln == 0) {
        s_bv = v; s_bi = ix;
        s_sel[it] = (v > NEG_INF) ? ix : -1;
      }
    }
    __syncthreads();
    float bv = s_bv;
    int   bi = s_bi;
    if (!(bv > NEG_INF)) break;  // uniform: nothing active remains

    // ---- fused: suppress by IoU vs best + rescan local argmax
    float4 q = cb[bi];  // canonicalized best box (broadcast L2 load)
    float areaQ = (q.y - q.x) * (q.w - q.z);
    lv = NEG_INF; li = t;
    for (int k = 0; k < NBOX / 1024; ++k) {
      int j = t + (k << 10);
      float sc = s_sc[j];
      if (sc > NEG_INF) {
        float4 kk = cb[j];
        float i0 = fmaxf(fminf(q.y, kk.y) - fmaxf(q.x, kk.x), 0.f);
        float i1 = fmaxf(fminf(q.w, kk.w) - fmaxf(q.z, kk.z), 0.f);
        float inter = i0 * i1;
        float areaK = (kk.y - kk.x) * (kk.w - kk.z);
        float uni = areaQ + areaK - inter;
        float iou = (uni > 0.f) ? (inter / uni) : 0.f;
        if (iou > IOU_THR) { sc = NEG_INF; s_sc[j] = sc; }
      }
      if (sc > lv) { lv = sc; li = j; }
    }
    // no barrier needed: each thread touches only its own s_sc slots;
    // s_rv/s_bv hazards are covered by the two barriers above.
  }
  __syncthreads();
  int* sb = sel + b * MAX_DET;
  for (int i = t; i < MAX_DET; i += 1024) sb[i] = s_sel[i];
}

// ---------------------------------------------------------------------------
// Pass C: gather outputs. One 128-thread block per detection (3200 blocks).
// Recomputes softmax for the 3200 selected rows only (cheaper than storing
// per-row stats for all 512K rows).
// ---------------------------------------------------------------------------
__global__ void gather_kernel(const float* __restrict__ boxp,
                              const float* __restrict__ clsp,
                              const int* __restrict__ sel,
                              float* __restrict__ out_box,
                              float* __restrict__ out_cls,
                              float* __restrict__ out_raw,
                              float* __restrict__ out_valid) {
  const int d = blockIdx.x;          // b*MAX_DET + i
  const int b = d / MAX_DET;
  const int t = threadIdx.x;
  __shared__ float red[128];

  const int  s    = sel[d];
  const bool ok   = (s >= 0);
  const int  rowi = ok ? s : 0;      // reference: idx = max(sel,0), masked by vm
  const float vm  = ok ? 1.f : 0.f;

  const float* cp = clsp + ((size_t)b * NBOX + rowi) * NCLS;
  float x = (t < NCLS) ? cp[t] : NEG_INF;
  red[t] = x;
  __syncthreads();
#pragma unroll
  for (int o = 64; o > 0; o >>= 1) {
    if (t < o) red[t] = fmaxf(red[t], red[t + o]);
    __syncthreads();
  }
  float m = red[0];
  __syncthreads();
  float e = (t < NCLS) ? __expf(x - m) : 0.f;
  red[t] = e;
  __syncthreads();
#pragma unroll
  for (int o = 64; o > 0; o >>= 1) {
    if (t < o) red[t] += red[t + o];
    __syncthreads();
  }
  float inv = 1.f / red[0];

  if (t < NCLS) {
    out_cls[(size_t)d * NCLS + t] = e * inv * vm;
    out_raw[(size_t)d * NCLS + t] = x * vm;
  }
  if (t < 4) {
    const float* bp = boxp + ((size_t)b * NBOX + rowi) * 4;
    float bx = bp[0], by = bp[1], bw = bp[2] - bp[0], bh = bp[3] - bp[1];
    float val = (t == 0) ? bx : (t == 1) ? by : (t == 2) ? bw : bh;
    out_box[(size_t)d * 4 + t] = val * vm;
  }
  if (t == 0) out_valid[d] = vm;
}

// ---------------------------------------------------------------------------
extern "C" void kernel_launch(void* const* d_in, const int* in_sizes, int n_in,
                              void* d_out, int out_size, void* d_ws, size_t ws_size,
                              hipStream_t stream) {
  const float* boxp = (const float*)d_in[0];  // [16,32768,4]
  const float* clsp = (const float*)d_in[1];  // [16,32768,80]

  float* out      = (float*)d_out;            // tuple concatenated flat
  float* out_box  = out;                                          // 16*200*4
  float* out_cls  = out_box + BATCH * MAX_DET * 4;                // 16*200*80
  float* out_raw  = out_cls + (size_t)BATCH * MAX_DET * NCLS;     // 16*200*80
  float* out_val  = out_raw + (size_t)BATCH * MAX_DET * NCLS;     // 16*200

  char*   ws    = (char*)d_ws;
  float*  ms    = (float*)ws;                                               // 2 MB
  float4* canon = (float4*)(ws + (size_t)BATCH * NBOX * sizeof(float));     // 8 MB
  int*    sel   = (int*)(ws + (size_t)BATCH * NBOX * (sizeof(float) + sizeof(float4)));

  score_canon_kernel<<<(BATCH * NBOX) / 256, 256, 0, stream>>>(boxp, clsp, ms, canon);
  nms_kernel<<<BATCH, 1024, 0, stream>>>(ms, canon, sel);
  gather_kernel<<<BATCH * MAX_DET, 128, 0, stream>>>(boxp, clsp, sel,
                                                     out_box, out_cls, out_raw, out_val);
}